// LinearUpsampling3D_29661044146303
// MI455X (gfx1250) — compile-verified
//
#include <hip/hip_runtime.h>

// Trilinear x2 upsampling, [2,64,64,64,64] f32 -> [2,128,128,128,64] f32.
// Memory-bound: ~1.2 GB HBM traffic, ~52us floor at 23.3 TB/s.
// Strategy: per-(b,id,ih) workgroup stages 4 input rows (64KB) into LDS via
// CDNA5 async global->LDS b128 loads (ASYNCcnt), computes the 2x2x128x64
// output tile from LDS with shared w-taps, writes with non-temporal b128
// stores so the 1.07GB write-once stream doesn't evict the L2-resident input
// (134MB < 192MB L2).

typedef float v4f __attribute__((ext_vector_type(4)));
typedef int   v4i __attribute__((ext_vector_type(4)));
// async-to-LDS builtin wants typed, address-space-qualified int4 pointers
typedef __attribute__((address_space(1))) v4i* gv4i_p;   // global
typedef __attribute__((address_space(3))) v4i* lv4i_p;   // LDS

#ifndef __has_builtin
#define __has_builtin(x) 0
#endif

#if __has_builtin(__builtin_amdgcn_global_load_async_to_lds_b128)
#define USE_ASYNC_LDS 1
#else
#define USE_ASYNC_LDS 0
#endif

#define TPB 256

__global__ __launch_bounds__(TPB) void upsample3d_lin_kernel(
    const float* __restrict__ in, float* __restrict__ out)
{
  // Input  [B=2, D=64, H=64, W=64, C=64], channels-last, f32.
  // Output [B=2, 128, 128, 128, C=64].
  // 1-D rule per dim: out[2i+1] = x[i]; out[2i] = 0.5*(x[i-1]+x[i]), zero pad.
  constexpr int ROW_V4 = 64 * 16;          // one (d,h) row: 64 w * 16 v4f = 1024 v4f
  __shared__ v4f lds[4 * ROW_V4];          // 4 rows = 64 KB

  const int bid = blockIdx.x;              // 0 .. 2*64*64-1
  const int ih  = bid & 63;
  const int id  = (bid >> 6) & 63;
  const int b   = bid >> 12;
  const int tid = threadIdx.x;

  const v4f* gin = (const v4f*)in;

  // ---- Stage rows r = dIdx*2 + hIdx : (id-1+dIdx, ih-1+hIdx) ----
#pragma unroll
  for (int r = 0; r < 4; ++r) {
    const int d = id - 1 + (r >> 1);
    const int h = ih - 1 + (r & 1);
    v4f* lrow = &lds[r * ROW_V4];
    if (d < 0 || h < 0) {
      // OOB halo -> zeros (only ~3% of workgroups hit this)
#pragma unroll
      for (int e = tid; e < ROW_V4; e += TPB) {
        v4f z = {0.0f, 0.0f, 0.0f, 0.0f};
        lrow[e] = z;
      }
    } else {
      const v4f* grow = gin + (size_t)((b * 64 + d) * 64 + h) * ROW_V4;
#pragma unroll
      for (int e = tid; e < ROW_V4; e += TPB) {
#if USE_ASYNC_LDS
        __builtin_amdgcn_global_load_async_to_lds_b128(
            (gv4i_p)(grow + e),
            (lv4i_p)(lrow + e),
            /*offset=*/0, /*cpol=*/0);
#else
        lrow[e] = grow[e];
#endif
      }
    }
  }
#if USE_ASYNC_LDS
#if __has_builtin(__builtin_amdgcn_s_wait_asynccnt)
  __builtin_amdgcn_s_wait_asynccnt(0);
#else
  asm volatile("s_wait_asynccnt 0" ::: "memory");
#endif
#endif
  __syncthreads();

  // ---- Compute 2x2x128x64 output tile ----
  const int c4 = tid & 15;   // which float4 of the 64 channels
  const int a  = tid >> 4;   // 0..15 : low bits of ow
  v4f* gout = (v4f*)out;

  // Output row bases for the four (pd, ph) parity combos.
  const size_t ob00 = ((size_t)((b * 128 + 2 * id + 0) * 128 + (2 * ih + 0))) * (size_t)(128 * 16);
  const size_t ob01 = ((size_t)((b * 128 + 2 * id + 0) * 128 + (2 * ih + 1))) * (size_t)(128 * 16);
  const size_t ob10 = ((size_t)((b * 128 + 2 * id + 1) * 128 + (2 * ih + 0))) * (size_t)(128 * 16);
  const size_t ob11 = ((size_t)((b * 128 + 2 * id + 1) * 128 + (2 * ih + 1))) * (size_t)(128 * 16);

#pragma unroll
  for (int j = 0; j < 8; ++j) {
    const int ow = a + 16 * j;      // 0..127
    const int iw = ow >> 1;
    const int ci = iw * 16 + c4;    // v4f index within a row

    // w-taps per LDS row, shared by all four outputs:
    //   odd ow : t[r] = x_r[iw]
    //   even ow: t[r] = 0.5*(x_r[iw-1] + x_r[iw])   (x[-1] = 0)
    v4f t[4];
#pragma unroll
    for (int r = 0; r < 4; ++r) t[r] = lds[r * ROW_V4 + ci];
    if (!(ow & 1)) {
#pragma unroll
      for (int r = 0; r < 4; ++r) {
        v4f l = {0.0f, 0.0f, 0.0f, 0.0f};
        if (iw > 0) l = lds[r * ROW_V4 + ci - 16];
        t[r] = 0.5f * (t[r] + l);
      }
    }

    // Row-combination (d/h) weights, with shared partial sums:
    //   o00 = 0.25*(t0+t1+t2+t3); o01 = 0.5*(t1+t3); o10 = 0.5*(t2+t3); o11 = t3
    const v4f s13 = t[1] + t[3];
    const v4f s23 = t[2] + t[3];
    const v4f o00 = 0.25f * (t[0] + t[2] + s13);
    const v4f o01 = 0.5f * s13;
    const v4f o10 = 0.5f * s23;
    const v4f o11 = t[3];

    const size_t off = (size_t)ow * 16 + c4;
    __builtin_nontemporal_store(o00, gout + ob00 + off);
    __builtin_nontemporal_store(o01, gout + ob01 + off);
    __builtin_nontemporal_store(o10, gout + ob10 + off);
    __builtin_nontemporal_store(o11, gout + ob11 + off);
  }
}

extern "C" void kernel_launch(void* const* d_in, const int* in_sizes, int n_in,
                              void* d_out, int out_size, void* d_ws, size_t ws_size,
                              hipStream_t stream) {
  (void)in_sizes; (void)n_in; (void)out_size; (void)d_ws; (void)ws_size;
  const float* in = (const float*)d_in[0];
  float* out = (float*)d_out;
  dim3 grid(2 * 64 * 64);   // one workgroup per (b, id, ih)
  dim3 block(TPB);          // 8 waves (wave32)
  upsample3d_lin_kernel<<<grid, block, 0, stream>>>(in, out);
}